// KaliSetModule_11003706213140
// MI455X (gfx1250) — compile-verified
//
#include <hip/hip_runtime.h>

#define DIM     3
#define N_ITER  64
#define EPS     1e-5f
#define PPT     4                 // pixels per thread
#define BLOCK   256

#if __has_builtin(__builtin_amdgcn_global_load_async_to_lds_b32) && \
    __has_builtin(__builtin_amdgcn_s_wait_asynccnt)
#define KALI_ASYNC_LDS 1
#else
#define KALI_ASYNC_LDS 0
#endif

typedef __attribute__((address_space(1))) int g_int_t;   // global (AS1)
typedef __attribute__((address_space(3))) int l_int_t;   // LDS (AS3)

__device__ __forceinline__ float kali_clamp01(float v) {
    return fminf(fmaxf(v, 0.0f), 1.0f);
}

// One Kali iteration for one pixel. |x| folds to a source modifier; the single
// precise reciprocal is shared by the 3 channels (ref divides 3x by the same
// denominator -> <=1-2 ulp difference, 3x fewer divide sequences).
__device__ __forceinline__ void kali_step(float kx, float ky, float kz,
                                          float& x, float& y, float& z,
                                          float& ax, float& ay, float& az)
{
    float dot = __builtin_fmaf(x, x, __builtin_fmaf(y, y, z * z));
    float inv = 1.0f / (EPS + dot);
    float sx = __builtin_fabsf(x) * inv;
    float sy = __builtin_fabsf(y) * inv;
    float sz = __builtin_fabsf(z) * inv;
    ax += sx; ay += sy; az += sz;
    x = sx - kx; y = sy - ky; z = sz - kz;
}

__global__ __launch_bounds__(BLOCK) void kali_kernel(
    const float* __restrict__ start_space,
    const float* __restrict__ zoom_p,
    const float* __restrict__ pos_p,
    const float* __restrict__ kali_p,
    float* __restrict__ out,
    long long ntotal)                        // total float count (= pixels*3)
{
    __shared__ float s_kp[N_ITER * DIM];     // 768 B

    const int t = threadIdx.x;
#if KALI_ASYNC_LDS
    if (t < N_ITER * DIM) {
        __builtin_amdgcn_global_load_async_to_lds_b32(
            (g_int_t*)(void*)(kali_p + t),
            (l_int_t*)(void*)(&s_kp[t]),
            /*imm offset*/0, /*cpol*/0);
    }
    __builtin_amdgcn_s_wait_asynccnt(0);
#else
    if (t < N_ITER * DIM) s_kp[t] = kali_p[t];
#endif
    __syncthreads();

    const float zoom = zoom_p[0];
    const float px = pos_p[0], py = pos_p[1], pz = pos_p[2];

    const long long tid  = (long long)blockIdx.x * BLOCK + t;
    const long long base = tid * (PPT * DIM);          // flat float index
    const float sc = 1.0f / (float)N_ITER;

    if (base + PPT * DIM <= ntotal) {
        // ---- fast path: 4 pixels = 48 contiguous bytes = 3x float4 ----
        const float4* in4 = (const float4*)(start_space + base);
        float4 A = in4[0], B = in4[1], C = in4[2];

        float x0 = fmaf(A.x, zoom, px), y0 = fmaf(A.y, zoom, py), z0 = fmaf(A.z, zoom, pz);
        float x1 = fmaf(A.w, zoom, px), y1 = fmaf(B.x, zoom, py), z1 = fmaf(B.y, zoom, pz);
        float x2 = fmaf(B.z, zoom, px), y2 = fmaf(B.w, zoom, py), z2 = fmaf(C.x, zoom, pz);
        float x3 = fmaf(C.y, zoom, px), y3 = fmaf(C.z, zoom, py), z3 = fmaf(C.w, zoom, pz);

        float ax0=0.f,ay0=0.f,az0=0.f, ax1=0.f,ay1=0.f,az1=0.f;
        float ax2=0.f,ay2=0.f,az2=0.f, ax3=0.f,ay3=0.f,az3=0.f;

#pragma unroll 4
        for (int it = 0; it < N_ITER; ++it) {
            float kx = s_kp[it*3+0], ky = s_kp[it*3+1], kz = s_kp[it*3+2];
            kali_step(kx,ky,kz, x0,y0,z0, ax0,ay0,az0);
            kali_step(kx,ky,kz, x1,y1,z1, ax1,ay1,az1);
            kali_step(kx,ky,kz, x2,y2,z2, ax2,ay2,az2);
            kali_step(kx,ky,kz, x3,y3,z3, ax3,ay3,az3);
        }

        float4 O0 = { kali_clamp01(ax0*sc), kali_clamp01(ay0*sc),
                      kali_clamp01(az0*sc), kali_clamp01(ax1*sc) };
        float4 O1 = { kali_clamp01(ay1*sc), kali_clamp01(az1*sc),
                      kali_clamp01(ax2*sc), kali_clamp01(ay2*sc) };
        float4 O2 = { kali_clamp01(az2*sc), kali_clamp01(ax3*sc),
                      kali_clamp01(ay3*sc), kali_clamp01(az3*sc) };
        float4* o4 = (float4*)(out + base);
        o4[0] = O0; o4[1] = O1; o4[2] = O2;
    } else {
        // ---- cold tail: per-pixel scalar path (unused for 2048^2 but safe) ----
        for (int p = 0; p < PPT; ++p) {
            long long pb = base + (long long)p * DIM;
            if (pb + DIM > ntotal) break;
            float x = fmaf(start_space[pb+0], zoom, px);
            float y = fmaf(start_space[pb+1], zoom, py);
            float z = fmaf(start_space[pb+2], zoom, pz);
            float ax=0.f, ay=0.f, az=0.f;
            for (int it = 0; it < N_ITER; ++it) {
                kali_step(s_kp[it*3+0], s_kp[it*3+1], s_kp[it*3+2],
                          x, y, z, ax, ay, az);
            }
            out[pb+0] = kali_clamp01(ax*sc);
            out[pb+1] = kali_clamp01(ay*sc);
            out[pb+2] = kali_clamp01(az*sc);
        }
    }
}

extern "C" void kernel_launch(void* const* d_in, const int* in_sizes, int n_in,
                              void* d_out, int out_size, void* d_ws, size_t ws_size,
                              hipStream_t stream)
{
    (void)n_in; (void)d_ws; (void)ws_size;
    const float* start_space = (const float*)d_in[0];
    const float* zoom_p      = (const float*)d_in[1];
    const float* pos_p       = (const float*)d_in[2];
    const float* kali_p      = (const float*)d_in[3];
    float* out = (float*)d_out;

    const long long ntotal   = (long long)out_size;          // pixels * 3 floats
    const long long npix     = ntotal / DIM;
    const long long nthreads = (npix + PPT - 1) / PPT;
    const int blocks         = (int)((nthreads + BLOCK - 1) / BLOCK);

    kali_kernel<<<blocks, BLOCK, 0, stream>>>(start_space, zoom_p, pos_p,
                                              kali_p, out, ntotal);
}